// BasisGATLayer_15805479649714
// MI455X (gfx1250) — compile-verified
//
#include <hip/hip_runtime.h>
#include <hip/hip_bf16.h>

#define BB   16
#define SEQ  100
#define DD   128
#define HH   8
#define NSS  101
#define HDD  16
#define XP   132      // padded LDS pitch (132 % 64 = 4 -> conflict-free 16-lane column reads)
#define NEG_INF (-1e9f)

typedef __attribute__((ext_vector_type(2))) float v2f;
typedef __attribute__((ext_vector_type(8))) float v8f;

static __device__ __forceinline__ float lrelu(float x) { return x > 0.f ? x : 0.01f * x; }

// ---------------------------------------------------------------------------
// Kernel 1: per (b,n) row -- q/k/v projections, term_q/term_k scalars,
// and the factored edge-MLP first layer A = d@W1a.T, Bv = d@W1b.T.
// ---------------------------------------------------------------------------
__global__ __launch_bounds__(128)
void gat_proj_kernel(const float* __restrict__ desc, const float* __restrict__ nv,
                     const float* __restrict__ Wq, const float* __restrict__ bq,
                     const float* __restrict__ Wk, const float* __restrict__ bk,
                     const float* __restrict__ Wv, const float* __restrict__ bv,
                     const float* __restrict__ W1, const float* __restrict__ Wa,
                     float* __restrict__ v_ws, float* __restrict__ termq,
                     float* __restrict__ termk, float* __restrict__ A_ws,
                     float* __restrict__ B_ws)
{
    __shared__ float xs[DD];
    __shared__ float dsx[DD];

    const int blk = blockIdx.x;
    const int b = blk / NSS;
    const int n = blk % NSS;
    const int t = threadIdx.x;

    xs[t] = nv[((size_t)b * NSS + n) * DD + t];
    const bool hasd = (n < SEQ);
    if (hasd) dsx[t] = desc[((size_t)b * SEQ + n) * DD + t];
    __syncthreads();

    const float* wqr = Wq + (size_t)t * DD;
    const float* wkr = Wk + (size_t)t * DD;
    const float* wvr = Wv + (size_t)t * DD;
    float q = bq[t], k = bk[t], v = bv[t];
#pragma unroll 8
    for (int m = 0; m < DD; ++m) {
        const float x = xs[m];
        q += x * wqr[m];
        k += x * wkr[m];
        v += x * wvr[m];
    }
    v_ws[((size_t)b * NSS + n) * DD + t] = v;

    // term_q / term_k: leaky_relu then dot with Wa chunk, reduce over d (16 lanes)
    float lq = lrelu(q) * Wa[t & 15];
    float lk = lrelu(k) * Wa[16 + (t & 15)];
#pragma unroll
    for (int off = 1; off < 16; off <<= 1) {
        lq += __shfl_xor(lq, off, 16);
        lk += __shfl_xor(lk, off, 16);
    }
    if ((t & 15) == 0) {
        termq[((size_t)b * NSS + n) * HH + (t >> 4)] = lq;
        termk[((size_t)b * NSS + n) * HH + (t >> 4)] = lk;
    }

    if (hasd) {
        const float* w1r = W1 + (size_t)t * (2 * DD);
        float a = 0.f, bb2 = 0.f;
#pragma unroll 8
        for (int m = 0; m < DD; ++m) {
            const float x = dsx[m];
            a   += x * w1r[m];
            bb2 += x * w1r[DD + m];
        }
        A_ws[((size_t)b * SEQ + n) * DD + t] = a;
        B_ws[((size_t)b * SEQ + n) * DD + t] = bb2;
    }
}

// ---------------------------------------------------------------------------
// Kernel 2: one block per (b,i) attention row. 4 waves; each wave owns
// 16-pair j tiles: builds relu(LN(A_i+B_j+b1)) in LDS, WMMA f32 16x16x4 GEMM
// against W2, reduces term_e per head via LDS staging (no bpermute storms).
// Block then does softmax + ctx.
// ---------------------------------------------------------------------------
__global__ __launch_bounds__(128)
void gat_row_kernel(const float* __restrict__ mask_M,
                    const float* __restrict__ b1, const float* __restrict__ ln_g,
                    const float* __restrict__ ln_b, const float* __restrict__ W2,
                    const float* __restrict__ b2, const float* __restrict__ Wa,
                    const float* __restrict__ ba,
                    const float* __restrict__ v_ws, const float* __restrict__ termq,
                    const float* __restrict__ termk, const float* __restrict__ A_ws,
                    const float* __restrict__ B_ws,
                    float* __restrict__ out_basis, float* __restrict__ out_attn)
{
    __shared__ float Xs[4][16][XP];       // per-wave activation tiles (33.8 KB)
    __shared__ float red[4][16][17];      // per-wave epilogue reduction scratch
    __shared__ float logit_s[HH][112];    // term_e per (head, j)
    __shared__ float attn_s[HH][112];
    __shared__ float Arow[DD];
    __shared__ float b1s[DD], gs[DD], bs[DD], b2s[DD], was[48];

    const int blk = blockIdx.x;
    const int b = blk / NSS;
    const int i = blk % NSS;
    const int t = threadIdx.x;
    const int wave = t >> 5;
    const int lane = t & 31;
    const int half = lane >> 4;
    const int l16 = lane & 15;

    if (t < DD) {
        b1s[t] = b1[t]; gs[t] = ln_g[t]; bs[t] = ln_b[t]; b2s[t] = b2[t];
        if (i > 0) Arow[t] = A_ws[((size_t)b * SEQ + (i - 1)) * DD + t];
    }
    if (t < 48) was[t] = Wa[t];
    for (int j = t; j < HH * 112; j += 128) (&logit_s[0][0])[j] = 0.f;
    __syncthreads();

    // ---- per-wave j tiles ------------------------------------------------
    for (int tile = wave; tile < 7; tile += 4) {
        const int j0 = tile * 16;
        float (*X)[XP] = Xs[wave];

        // Build X[p][k] = relu(LN(A + B + b1)), all 16 pairs in parallel:
        // 2 lanes per pair, each owns 64 contiguous k's; one shfl_xor(1)
        // combines the pair's sum/sumsq.
        {
            const int p   = lane >> 1;
            const int sub = lane & 1;
            const int j   = j0 + p;
            const int kb  = sub * 64;
            if (j >= 1 && j < NSS) {
                const float* asrc = (i == 0) ? (A_ws + ((size_t)b * SEQ + (j - 1)) * DD)
                                             : Arow;
                const float* bsrc = B_ws + ((size_t)b * SEQ + (j - 1)) * DD;
                float s = 0.f, ss = 0.f;
#pragma unroll 8
                for (int kk = 0; kk < 64; ++kk) {
                    const float x = asrc[kb + kk] + bsrc[kb + kk] + b1s[kb + kk];
                    s += x;
                    ss += x * x;
                }
                s  += __shfl_xor(s,  1, 32);
                ss += __shfl_xor(ss, 1, 32);
                const float mu  = s * (1.f / 128.f);
                const float var = ss * (1.f / 128.f) - mu * mu;
                const float rs  = rsqrtf(var + 1e-5f);
#pragma unroll 8
                for (int kk = 0; kk < 64; ++kk) {
                    const float x = asrc[kb + kk] + bsrc[kb + kk] + b1s[kb + kk];
                    X[p][kb + kk] = fmaxf(0.f, (x - mu) * rs * gs[kb + kk] + bs[kb + kk]);
                }
            } else {
#pragma unroll 8
                for (int kk = 0; kk < 64; ++kk) X[p][kb + kk] = 0.f;
            }
        }
        // wave-local LDS visibility before cross-lane WMMA fragment reads
        asm volatile("s_wait_dscnt 0" ::: "memory");

        // GEMM: e[16 x 128] = X[16 x 128] @ W2^T, one n-tile (=one head) at a time
        for (int n0 = 0; n0 < 8; ++n0) {
            v8f acc = {};
            const float* w2base = W2 + (size_t)(n0 * 16 + l16) * DD + half * 2;
#pragma unroll 4
            for (int k0 = 0; k0 < DD; k0 += 4) {
                v2f a, bf;
                a.x  = X[l16][k0 + half * 2 + 0];
                a.y  = X[l16][k0 + half * 2 + 1];
                bf.x = w2base[k0 + 0];
                bf.y = w2base[k0 + 1];
                acc = __builtin_amdgcn_wmma_f32_16x16x4_f32(
                        false, a, false, bf, (short)0, acc, false, false);
            }
            // term_e epilogue: stage lr(e + b2) * Wa into LDS, then lanes 0-15
            // each sum one contiguous 16-float row. D-layout: VGPR r, lanes
            // 0-15 -> (M=r, N=lane); lanes 16-31 -> (M=r+8, N=lane-16).
#pragma unroll
            for (int r = 0; r < 8; ++r) {
                const float e = acc[r] + b2s[n0 * 16 + l16];
                red[wave][r + half * 8][l16] = lrelu(e) * was[32 + l16];
            }
            asm volatile("s_wait_dscnt 0" ::: "memory");
            if (lane < 16) {
                const float* row = red[wave][lane];
                float s0 = 0.f;
#pragma unroll
                for (int n = 0; n < 16; ++n) s0 += row[n];
                logit_s[n0][j0 + lane] = s0;
            }
            asm volatile("s_wait_dscnt 0" ::: "memory");
        }
    }
    __syncthreads();

    // ---- softmax + attn + ctx: 8 groups of 16 threads (h = t/16, d = t%16)
    const int h = t >> 4;
    const int d = t & 15;
    const float tq  = termq[((size_t)b * NSS + i) * HH + h];
    const float bav = ba[0];

    float l[7];
    float mx = NEG_INF;
#pragma unroll
    for (int c = 0; c < 7; ++c) {
        const int j = d + c * 16;
        float lv = NEG_INF;
        const bool masked = (j == 0) || (j >= NSS) || (i >= 1 && j == i);
        if (!masked) {
            lv = tq + termk[((size_t)b * NSS + j) * HH + h] + logit_s[h][j] + bav;
            if (i >= 1) {
                float pm = mask_M[(((size_t)b * HH + h) * SEQ + (i - 1)) * SEQ + (j - 1)];
                pm = fminf(fmaxf(pm, 1e-6f), 1.f - 1e-6f);
                lv += logf(pm) - log1pf(-pm);       // GAMMA = 1
            }
        }
        l[c] = lv;
        mx = fmaxf(mx, lv);
    }
#pragma unroll
    for (int off = 1; off < 16; off <<= 1) mx = fmaxf(mx, __shfl_xor(mx, off, 16));

    float s = 0.f;
#pragma unroll
    for (int c = 0; c < 7; ++c) {
        const float e = (l[c] > -1e8f) ? expf(l[c] - mx) : 0.f;
        l[c] = e;
        s += e;
    }
#pragma unroll
    for (int off = 1; off < 16; off <<= 1) s += __shfl_xor(s, off, 16);
    const float inv = 1.f / s;

#pragma unroll
    for (int c = 0; c < 7; ++c) {
        const int j = d + c * 16;
        if (j < NSS) {
            const float a = l[c] * inv;
            attn_s[h][j] = a;
            out_attn[(((size_t)b * HH + h) * NSS + i) * NSS + j] = a;
        }
    }
    __syncthreads();

    // ctx[b,h,i,d] = sum_j attn * v ; basis = ctx transposed to [B,NS,H,HD]
    float acc = 0.f;
    for (int j = 1; j < NSS; ++j)
        acc += attn_s[h][j] * v_ws[((size_t)b * NSS + j) * DD + h * HDD + d];
    out_basis[(((size_t)b * NSS + i) * HH + h) * HDD + d] = acc;
}

// ---------------------------------------------------------------------------
extern "C" void kernel_launch(void* const* d_in, const int* in_sizes, int n_in,
                              void* d_out, int out_size, void* d_ws, size_t ws_size,
                              hipStream_t stream) {
    const float* desc = (const float*)d_in[0];
    const float* nv   = (const float*)d_in[1];
    const float* mask = (const float*)d_in[2];
    const float* Wq   = (const float*)d_in[3];
    const float* bq   = (const float*)d_in[4];
    const float* Wk   = (const float*)d_in[5];
    const float* bk   = (const float*)d_in[6];
    const float* Wv   = (const float*)d_in[7];
    const float* bv   = (const float*)d_in[8];
    const float* W1   = (const float*)d_in[9];
    const float* b1   = (const float*)d_in[10];
    const float* ln_g = (const float*)d_in[11];
    const float* ln_b = (const float*)d_in[12];
    const float* W2   = (const float*)d_in[13];
    const float* b2   = (const float*)d_in[14];
    const float* Wa   = (const float*)d_in[15];
    const float* ba   = (const float*)d_in[16];

    float* ws     = (float*)d_ws;
    float* v_ws   = ws;                              // B*NS*D   = 206848
    float* termq  = v_ws  + (size_t)BB * NSS * DD;   // B*NS*H   = 12928
    float* termk  = termq + (size_t)BB * NSS * HH;
    float* A_ws   = termk + (size_t)BB * NSS * HH;   // B*SEQ*D  = 204800
    float* B_ws   = A_ws  + (size_t)BB * SEQ * DD;

    float* out_basis = (float*)d_out;                       // B*NS*H*HD
    float* out_attn  = out_basis + (size_t)BB * NSS * HH * HDD;

    const int nblk = BB * NSS;
    gat_proj_kernel<<<nblk, 128, 0, stream>>>(desc, nv, Wq, bq, Wk, bk, Wv, bv,
                                              W1, Wa, v_ws, termq, termk, A_ws, B_ws);
    gat_row_kernel<<<nblk, 128, 0, stream>>>(mask, b1, ln_g, ln_b, W2, b2, Wa, ba,
                                             v_ws, termq, termk, A_ws, B_ws,
                                             out_basis, out_attn);
}